// SplineDepthHeadV3_25615184954154
// MI455X (gfx1250) — compile-verified
//
#include <hip/hip_runtime.h>
#include <hip/hip_bf16.h>

typedef float v2f __attribute__((ext_vector_type(2)));
typedef float v8f __attribute__((ext_vector_type(8)));

// ---------------------------------------------------------------------------
// Batched GEMM, specialized for M == 64:
//   C[z] = act(A[z](64xK) @ B[z](KxN) + bias[z](N))
// One wave32 computes a 64x16 output slab (four 16x16 WMMA tiles) so each
// B element is loaded exactly ONCE (weights dominate GEMM traffic here).
// blockDim.x = 128 (4 waves -> 4 consecutive N-tiles). grid = (ceil(N/64), 1, Z)
// VGPR layouts per CDNA5 ISA 7.12.2:
//   A 16x4 : lane L<16 -> row M=L, K={0,1}; lane L>=16 -> row M=L-16, K={2,3}
//   B 4x16 : lane L<16 -> col N=L, K={0,1}; lane L>=16 -> col N=L-16, K={2,3}
//   C/D    : vgpr i, lanes 0-15 -> M=i, N=lane; lanes 16-31 -> M=i+8, N=lane-16
// ---------------------------------------------------------------------------
__global__ void gemm_wmma_f32_kernel(const float* __restrict__ A,
                                     const float* __restrict__ Bm,
                                     const float* __restrict__ bias,
                                     float* __restrict__ C,
                                     int M /* == 64 */, int N, int K,
                                     long sA, long sB, long sBias, long sC,
                                     int act_leaky)
{
    const int z = blockIdx.z;
    A  += (long)z * sA;
    Bm += (long)z * sB;
    C  += (long)z * sC;
    const float* bptr = bias ? (bias + (long)z * sBias) : nullptr;

    const int wave  = threadIdx.x >> 5;          // wave-uniform
    const int ntile = blockIdx.x * 4 + wave;
    const int n0    = ntile * 16;
    if (n0 >= N) return;                          // uniform per wave: EXEC stays all-1

    const int lane = threadIdx.x & 31;
    const int half = lane >> 4;                   // 0: K={0,1}, 1: K={2,3}
    const int col  = lane & 15;

    int bcol = n0 + col; if (bcol >= N) bcol = N - 1;   // clamp partial N tile (no OOB)

    const float* Ap = A + (long)col * K + 2 * half;     // row (mt*16+col), contiguous K pair
    const float* Bp = Bm + bcol + (long)(2 * half) * N;

    v8f acc0 = {}, acc1 = {}, acc2 = {}, acc3 = {};
#pragma unroll 2
    for (int k0 = 0; k0 < K; k0 += 4) {
        v2f b;
        b.x = Bp[(long)k0 * N];                          // B[k0+2h  ][bcol]
        b.y = Bp[(long)k0 * N + N];                      // B[k0+2h+1][bcol]
        v2f a0 = *(const v2f*)(Ap + k0);                 // rows  0..15
        v2f a1 = *(const v2f*)(Ap + (long)16 * K + k0);  // rows 16..31
        v2f a2 = *(const v2f*)(Ap + (long)32 * K + k0);  // rows 32..47
        v2f a3 = *(const v2f*)(Ap + (long)48 * K + k0);  // rows 48..63
        acc0 = __builtin_amdgcn_wmma_f32_16x16x4_f32(false, a0, false, b, (short)0, acc0, false, false);
        acc1 = __builtin_amdgcn_wmma_f32_16x16x4_f32(false, a1, false, b, (short)0, acc1, false, false);
        acc2 = __builtin_amdgcn_wmma_f32_16x16x4_f32(false, a2, false, b, (short)0, acc2, false, false);
        acc3 = __builtin_amdgcn_wmma_f32_16x16x4_f32(false, a3, false, b, (short)0, acc3, false, false);
    }

    const int nrem = N - n0;
    const float bv = (bptr && col < nrem) ? bptr[n0 + col] : 0.0f;
    v8f accs[4] = {acc0, acc1, acc2, acc3};
#pragma unroll
    for (int mt = 0; mt < 4; ++mt) {
#pragma unroll
        for (int i = 0; i < 8; ++i) {
            int m = mt * 16 + i + 8 * half;               // < 64 == M
            if (col < nrem) {
                float v = accs[mt][i] + bv;
                if (act_leaky) v = (v >= 0.0f) ? v : 0.2f * v;
                C[(long)m * N + n0 + col] = v;
            }
        }
    }
}

// ---------------------------------------------------------------------------
// effective bias for rank MLP layer 1:
//   effb1[r][j] = rm_b1[r][j] + sum_d rank_emb[r][d] * rm_w1[r][256+d][j]
// ---------------------------------------------------------------------------
__global__ void effb1_kernel(const float* __restrict__ rank_emb,
                             const float* __restrict__ rm_w1,
                             const float* __restrict__ rm_b1,
                             float* __restrict__ effb1)
{
    int idx = blockIdx.x * blockDim.x + threadIdx.x;   // 24*256
    if (idx >= 24 * 256) return;
    int r = idx >> 8, j = idx & 255;
    const float* e = rank_emb + r * 256;
    const float* w = rm_w1 + (long)r * 512 * 256 + 256 * 256 + j;
    float s = rm_b1[idx];
#pragma unroll 4
    for (int d = 0; d < 256; ++d) s = fmaf(e[d], w[(long)d * 256], s);
    effb1[idx] = s;
}

// ---------------------------------------------------------------------------
// tiny softmaxes of mult_w(16), addx_w(8), addy_w(8)
// ---------------------------------------------------------------------------
__device__ __forceinline__ void softmax_n(const float* in, int n, float* out)
{
    float mx = -1e30f;
    for (int i = 0; i < n; ++i) mx = fmaxf(mx, in[i]);
    float s = 0.0f;
    for (int i = 0; i < n; ++i) s += expf(in[i] - mx);
    float inv = 1.0f / s;
    for (int i = 0; i < n; ++i) out[i] = expf(in[i] - mx) * inv;
}

__global__ void softmax_prep_kernel(const float* mw, const float* axw, const float* ayw,
                                    float* wm, float* wax, float* way)
{
    if (blockIdx.x == 0 && threadIdx.x == 0) {
        softmax_n(mw, 16, wm);
        softmax_n(axw, 8, wax);
        softmax_n(ayw, 8, way);
    }
}

// ---------------------------------------------------------------------------
// Cubic Hermite spline, matching reference _spline (N=24 cp, 512 outputs)
// ---------------------------------------------------------------------------
__device__ __forceinline__ float spline_slope(const float* cp, int k)
{
    const float c = 0.5f * (1.0f / 23.0f);
    if (k == 0)  return (cp[1]  - cp[0])  * c;
    if (k == 23) return (cp[23] - cp[22]) * c;
    return (cp[k + 1] - cp[k - 1]) * c;
}

__device__ __forceinline__ float spline_eval(const float* __restrict__ cp, int i)
{
    const float eps = 0.001f;
    float t  = eps + (1.0f - 2.0f * eps) * ((float)i * (1.0f / 511.0f));
    float ts = t * 23.0f;
    int seg = (int)floorf(ts);
    seg = min(max(seg, 0), 22);
    float tau = ts - (float)seg;
    tau = fminf(fmaxf(tau, 0.0f), 0.9999f);
    float t2 = tau * tau, t3 = t2 * tau;
    float h00 =  2.0f * t3 - 3.0f * t2 + 1.0f;
    float h10 =         t3 - 2.0f * t2 + tau;
    float h01 = -2.0f * t3 + 3.0f * t2;
    float h11 =         t3 -        t2;
    int s1 = min(seg + 1, 23);
    float pk  = cp[seg], pk1 = cp[s1];
    float mk  = spline_slope(cp, seg);
    float mk1 = spline_slope(cp, s1);
    return h00 * pk + h10 * mk + h01 * pk1 + h11 * mk1;
}

// u/v: out[(b*R + r)*512 + i] = w[r] * spline(P[r][b], i)   (P layout: (R,64,24))
__global__ void spline_uv_kernel(const float* __restrict__ P,
                                 const float* __restrict__ wvec,
                                 float* __restrict__ out, int R)
{
    int idx = blockIdx.x * blockDim.x + threadIdx.x;
    int total = 64 * R * 512;
    if (idx >= total) return;
    int i  = idx & 511;
    int br = idx >> 9;
    int r  = br % R;
    int b  = br / R;
    const float* cp = P + ((long)r * 64 + b) * 24;
    float w = wvec ? wvec[r] : 1.0f;
    out[idx] = w * spline_eval(cp, i);
}

// dx/dy: out[b*512+i] = sum_{r<8} w[r] * spline(P[r][b], i)
__global__ void spline_reduce_kernel(const float* __restrict__ P,
                                     const float* __restrict__ wvec,
                                     float* __restrict__ out)
{
    int idx = blockIdx.x * blockDim.x + threadIdx.x;   // 64*512
    if (idx >= 64 * 512) return;
    int i = idx & 511, b = idx >> 9;
    float s = 0.0f;
#pragma unroll
    for (int r = 0; r < 8; ++r)
        s = fmaf(wvec[r], spline_eval(P + ((long)r * 64 + b) * 24, i), s);
    out[idx] = s;
}

// ---------------------------------------------------------------------------
// Final fused kernel (store-bandwidth bound, 67 MB out):
// depth[b,h,w] = sum_r u'[b,r,w]*v[b,r,h] + dx[b,w] + dy[b,h] + gbias
// grid (64, 8), block 512: u' per-thread in 16 regs, v 16x64 tile in LDS.
// ---------------------------------------------------------------------------
__global__ void depth_final_kernel(const float* __restrict__ u,
                                   const float* __restrict__ v,
                                   const float* __restrict__ dx,
                                   const float* __restrict__ dy,
                                   const float* __restrict__ gbias,
                                   float* __restrict__ out)
{
    __shared__ float vs[16 * 64];
    __shared__ float dys[64];

    const int b  = blockIdx.x;
    const int h0 = blockIdx.y * 64;
    const int w  = threadIdx.x;   // 512

    for (int t = threadIdx.x; t < 16 * 64; t += 512) {
        int r = t >> 6, hh = t & 63;
        vs[t] = v[((long)b * 16 + r) * 512 + h0 + hh];
    }
    if (threadIdx.x < 64) dys[threadIdx.x] = dy[b * 512 + h0 + threadIdx.x];
    __syncthreads();

    float ur[16];
#pragma unroll
    for (int r = 0; r < 16; ++r) ur[r] = u[((long)b * 16 + r) * 512 + w];
    const float base = dx[b * 512 + w] + gbias[0];

    float* op = out + ((long)b * 512 + h0) * 512 + w;
    for (int hh = 0; hh < 64; ++hh) {
        float acc = base + dys[hh];
#pragma unroll
        for (int r = 0; r < 16; ++r) acc = fmaf(ur[r], vs[r * 64 + hh], acc);
        op[(long)hh * 512] = acc;
    }
}

// ---------------------------------------------------------------------------
extern "C" void kernel_launch(void* const* d_in, const int* in_sizes, int n_in,
                              void* d_out, int out_size, void* d_ws, size_t ws_size,
                              hipStream_t stream)
{
    const float* h        = (const float*)d_in[0];
    const float* rank_emb = (const float*)d_in[1];
    const float* st_w1    = (const float*)d_in[2];
    const float* st_b1    = (const float*)d_in[3];
    const float* st_w2    = (const float*)d_in[4];
    const float* st_b2    = (const float*)d_in[5];
    const float* rm_w1    = (const float*)d_in[6];
    const float* rm_b1    = (const float*)d_in[7];
    const float* rm_w2    = (const float*)d_in[8];
    const float* rm_b2    = (const float*)d_in[9];
    const float* mx_w1    = (const float*)d_in[10];
    const float* mx_b1    = (const float*)d_in[11];
    const float* mx_w2    = (const float*)d_in[12];
    const float* mx_b2    = (const float*)d_in[13];
    const float* my_w1    = (const float*)d_in[14];
    const float* my_b1    = (const float*)d_in[15];
    const float* my_w2    = (const float*)d_in[16];
    const float* my_b2    = (const float*)d_in[17];
    const float* ax_w1    = (const float*)d_in[18];
    const float* ax_b1    = (const float*)d_in[19];
    const float* ax_w2    = (const float*)d_in[20];
    const float* ax_b2    = (const float*)d_in[21];
    const float* ay_w1    = (const float*)d_in[22];
    const float* ay_b1    = (const float*)d_in[23];
    const float* ay_w2    = (const float*)d_in[24];
    const float* ay_b2    = (const float*)d_in[25];
    const float* mult_w   = (const float*)d_in[26];
    const float* addx_w   = (const float*)d_in[27];
    const float* addy_w   = (const float*)d_in[28];
    const float* gbias    = (const float*)d_in[29];
    float* out = (float*)d_out;

    float* ws = (float*)d_ws;
    // workspace layout (float offsets)
    float* ws_h1    = ws;                        // 64*512
    float* ws_sh    = ws_h1   + 64 * 512;        // 64*256
    float* ws_eb1   = ws_sh   + 64 * 256;        // 24*256
    float* ws_rfh   = ws_eb1  + 24 * 256;        // 24*64*256
    float* ws_rf    = ws_rfh  + 24 * 64 * 256;   // 24*64*256
    float* ws_gh    = ws_rf   + 24 * 64 * 256;   // 16*64*256 (reused across heads)
    float* ws_Pxm   = ws_gh   + 16 * 64 * 256;   // 16*64*24
    float* ws_Pym   = ws_Pxm  + 16 * 64 * 24;    // 16*64*24
    float* ws_Pxa   = ws_Pym  + 16 * 64 * 24;    // 8*64*24
    float* ws_Pya   = ws_Pxa  + 8 * 64 * 24;     // 8*64*24
    float* ws_u     = ws_Pya  + 8 * 64 * 24;     // 64*16*512
    float* ws_v     = ws_u    + 64 * 16 * 512;   // 64*16*512
    float* ws_dx    = ws_v    + 64 * 16 * 512;   // 64*512
    float* ws_dy    = ws_dx   + 64 * 512;        // 64*512
    float* ws_wm    = ws_dy   + 64 * 512;        // 16
    float* ws_wax   = ws_wm   + 16;              // 8
    float* ws_way   = ws_wax  + 8;               // 8

    dim3 blk128(128);

    // 0) softmaxes + effective rank bias
    softmax_prep_kernel<<<1, 32, 0, stream>>>(mult_w, addx_w, addy_w, ws_wm, ws_wax, ws_way);
    effb1_kernel<<<24, 256, 0, stream>>>(rank_emb, rm_w1, rm_b1, ws_eb1);

    // 1) shared MLP: hidden1 = leaky(h @ st_w1 + st_b1)  (64x768 @ 768x512)
    gemm_wmma_f32_kernel<<<dim3(8, 1, 1), blk128, 0, stream>>>(
        h, st_w1, st_b1, ws_h1, 64, 512, 768, 0, 0, 0, 0, 1);
    //    shared = hidden1 @ st_w2 + st_b2                (64x512 @ 512x256)
    gemm_wmma_f32_kernel<<<dim3(4, 1, 1), blk128, 0, stream>>>(
        ws_h1, st_w2, st_b2, ws_sh, 64, 256, 512, 0, 0, 0, 0, 0);

    // 2) rank MLPs (24 batched): rfh = leaky(shared @ rm_w1[:, :256, :] + effb1)
    gemm_wmma_f32_kernel<<<dim3(4, 1, 24), blk128, 0, stream>>>(
        ws_sh, rm_w1, ws_eb1, ws_rfh, 64, 256, 256,
        0, (long)512 * 256, 256, (long)64 * 256, 1);
    //    rf = rfh @ rm_w2 + rm_b2
    gemm_wmma_f32_kernel<<<dim3(4, 1, 24), blk128, 0, stream>>>(
        ws_rfh, rm_w2, rm_b2, ws_rf, 64, 256, 256,
        (long)64 * 256, (long)256 * 256, 256, (long)64 * 256, 0);

    const float* fm = ws_rf;                       // ranks 0..15
    const float* fa = ws_rf + (long)16 * 64 * 256; // ranks 16..23

    // 3) head mx: gh = leaky(fm @ mx_w1 + mx_b1); Pxm = gh @ mx_w2 + mx_b2
    gemm_wmma_f32_kernel<<<dim3(4, 1, 16), blk128, 0, stream>>>(
        fm, mx_w1, mx_b1, ws_gh, 64, 256, 256,
        (long)64 * 256, (long)256 * 256, 256, (long)64 * 256, 1);
    gemm_wmma_f32_kernel<<<dim3(1, 1, 16), blk128, 0, stream>>>(
        ws_gh, mx_w2, mx_b2, ws_Pxm, 64, 24, 256,
        (long)64 * 256, (long)256 * 24, 24, (long)64 * 24, 0);
    // head my
    gemm_wmma_f32_kernel<<<dim3(4, 1, 16), blk128, 0, stream>>>(
        fm, my_w1, my_b1, ws_gh, 64, 256, 256,
        (long)64 * 256, (long)256 * 256, 256, (long)64 * 256, 1);
    gemm_wmma_f32_kernel<<<dim3(1, 1, 16), blk128, 0, stream>>>(
        ws_gh, my_w2, my_b2, ws_Pym, 64, 24, 256,
        (long)64 * 256, (long)256 * 24, 24, (long)64 * 24, 0);
    // head ax
    gemm_wmma_f32_kernel<<<dim3(4, 1, 8), blk128, 0, stream>>>(
        fa, ax_w1, ax_b1, ws_gh, 64, 256, 256,
        (long)64 * 256, (long)256 * 256, 256, (long)64 * 256, 1);
    gemm_wmma_f32_kernel<<<dim3(1, 1, 8), blk128, 0, stream>>>(
        ws_gh, ax_w2, ax_b2, ws_Pxa, 64, 24, 256,
        (long)64 * 256, (long)256 * 24, 24, (long)64 * 24, 0);
    // head ay
    gemm_wmma_f32_kernel<<<dim3(4, 1, 8), blk128, 0, stream>>>(
        fa, ay_w1, ay_b1, ws_gh, 64, 256, 256,
        (long)64 * 256, (long)256 * 256, 256, (long)64 * 256, 1);
    gemm_wmma_f32_kernel<<<dim3(1, 1, 8), blk128, 0, stream>>>(
        ws_gh, ay_w2, ay_b2, ws_Pya, 64, 24, 256,
        (long)64 * 256, (long)256 * 24, 24, (long)64 * 24, 0);

    // 4) splines: u (wm folded in), v, dx, dy
    spline_uv_kernel<<<2048, 256, 0, stream>>>(ws_Pxm, ws_wm, ws_u, 16);
    spline_uv_kernel<<<2048, 256, 0, stream>>>(ws_Pym, nullptr, ws_v, 16);
    spline_reduce_kernel<<<128, 256, 0, stream>>>(ws_Pxa, ws_wax, ws_dx);
    spline_reduce_kernel<<<128, 256, 0, stream>>>(ws_Pya, ws_way, ws_dy);

    // 5) fused rank-16 outer-product accumulation + adds (67 MB streaming store)
    depth_final_kernel<<<dim3(64, 8), dim3(512), 0, stream>>>(
        ws_u, ws_v, ws_dx, ws_dy, gbias, out);
}